// NeuralDictionaryV15_38594576121970
// MI455X (gfx1250) — compile-verified
//
#include <hip/hip_runtime.h>
#include <math.h>

typedef __attribute__((ext_vector_type(2))) float v2f;
typedef __attribute__((ext_vector_type(8))) float v8f;

#define NROWS 262144
#define DDIM 512
#define VDIM 512
#define GRID 1024
#define TPB 256
#define WAVES (TPB / 32)
#define ROWS_PER_BLOCK (NROWS / GRID)        // 256
#define TILES_PER_BLOCK (ROWS_PER_BLOCK / 16) // 16
#define TILES_PER_WAVE (TILES_PER_BLOCK / WAVES) // 2

// Online-softmax combine: state (m, s=sum exp(l-m), idx of max)
__device__ __forceinline__ void online_combine(float& m, float& s, int& idx,
                                               float m2, float s2, int i2) {
  if (m2 > m) {
    s = s * __expf(m - m2) + s2;
    m = m2;
    idx = i2;
  } else {
    s = s + s2 * __expf(m2 - m);
  }
}

__global__ __launch_bounds__(TPB) void nd_phase1(const float* __restrict__ q,
                                                 const float* __restrict__ keys,
                                                 float4* __restrict__ partials) {
  __shared__ __align__(16) float qs[DDIM];
  __shared__ float sm[WAVES];
  __shared__ float ss[WAVES];
  __shared__ int   si[WAVES];

  // Stage query into LDS once (2 KB); reused 128x per tile per wave.
  for (int i = threadIdx.x; i < DDIM; i += TPB) qs[i] = q[i];
  __syncthreads();

  const int lane    = threadIdx.x & 31;
  const int wave    = threadIdx.x >> 5;
  const int halfSel = lane >> 4;      // which half-wave (K-slot group)
  const int ko      = halfSel * 2;    // K offset within group of 4
  const int mrow    = lane & 15;      // A-matrix row (M) this lane feeds

  float m = -__builtin_inff();
  float s = 0.0f;
  int   idx = 0;

  const int tile0 = blockIdx.x * TILES_PER_BLOCK + wave * TILES_PER_WAVE;

  for (int t = 0; t < TILES_PER_WAVE; ++t) {
    const int rowBase = (tile0 + t) * 16;
    const float* kp = keys + (size_t)(rowBase + mrow) * DDIM + ko;

    v8f c0 = {};
    v8f c1 = {};
#pragma unroll 4
    for (int kb = 0; kb < DDIM; kb += 8) {
      // A fragments: per-lane float2 of this lane's key row at its K slots
      v2f a0 = *(const v2f*)(kp + kb);
      v2f a1 = *(const v2f*)(kp + kb + 4);
      // B fragments: query chunk broadcast to all 16 columns (same K slots)
      v2f b0 = *(const v2f*)(&qs[kb + ko]);
      v2f b1 = *(const v2f*)(&qs[kb + 4 + ko]);
      c0 = __builtin_amdgcn_wmma_f32_16x16x4_f32(false, a0, false, b0,
                                                 (short)0, c0, false, false);
      c1 = __builtin_amdgcn_wmma_f32_16x16x4_f32(false, a1, false, b1,
                                                 (short)0, c1, false, false);
    }

    // D layout: VGPR r = logit of row (rowBase + halfSel*8 + r); every column
    // (lane within half-wave) holds the same value.
#pragma unroll
    for (int r = 0; r < 8; ++r) {
      float l = c0[r] + c1[r];
      int row = rowBase + halfSel * 8 + r;
      if (l > m) {
        s = s * __expf(m - l) + 1.0f;
        m = l;
        idx = row;
      } else {
        s += __expf(l - m);
      }
    }
  }

  // Lanes 0-15 carry rows 0-7 state, lanes 16-31 rows 8-15; one xor-16
  // exchange merges the two disjoint halves (duplicates merge with duplicates).
  {
    float m2 = __shfl_xor(m, 16, 32);
    float s2 = __shfl_xor(s, 16, 32);
    int   i2 = __shfl_xor(idx, 16, 32);
    online_combine(m, s, idx, m2, s2, i2);
  }

  if (lane == 0) { sm[wave] = m; ss[wave] = s; si[wave] = idx; }
  __syncthreads();

  if (threadIdx.x == 0) {
    float M = sm[0], S = ss[0];
    int   I = si[0];
    for (int w = 1; w < WAVES; ++w) online_combine(M, S, I, sm[w], ss[w], si[w]);
    partials[blockIdx.x] = make_float4(M, S, __int_as_float(I), 0.0f);
  }
}

__global__ __launch_bounds__(256) void nd_phase2(const float4* __restrict__ partials,
                                                 const float* __restrict__ values,
                                                 float* __restrict__ out) {
  __shared__ float sm[256];
  __shared__ float ss[256];
  __shared__ int   si[256];

  const int tid = threadIdx.x;
  float m = -__builtin_inff();
  float s = 0.0f;
  int   idx = 0;

  for (int i = tid; i < GRID; i += 256) {
    float4 p = partials[i];
    online_combine(m, s, idx, p.x, p.y, __float_as_int(p.z));
  }
  sm[tid] = m; ss[tid] = s; si[tid] = idx;
  __syncthreads();

  for (int off = 128; off > 0; off >>= 1) {
    if (tid < off) {
      float M = sm[tid], S = ss[tid];
      int   I = si[tid];
      online_combine(M, S, I, sm[tid + off], ss[tid + off], si[tid + off]);
      sm[tid] = M; ss[tid] = S; si[tid] = I;
    }
    __syncthreads();
  }

  const float S = ss[0];
  const int   I = si[0];
  const float inv = 1.0f / S;  // attention at argmax = exp(M-M)/S = 1/S
  for (int v = tid; v < VDIM; v += 256)
    out[v] = values[(size_t)I * VDIM + v] * inv;
}

extern "C" void kernel_launch(void* const* d_in, const int* in_sizes, int n_in,
                              void* d_out, int out_size, void* d_ws, size_t ws_size,
                              hipStream_t stream) {
  const float* query  = (const float*)d_in[0];
  const float* keys   = (const float*)d_in[1];
  const float* values = (const float*)d_in[2];
  float4* partials = (float4*)d_ws;  // GRID * 16 B = 16 KB of scratch

  nd_phase1<<<GRID, TPB, 0, stream>>>(query, keys, partials);
  nd_phase2<<<1, 256, 0, stream>>>(partials, values, (float*)d_out);
}